// HashEmbedderOptimized_49520972923487
// MI455X (gfx1250) — compile-verified
//
#include <hip/hip_runtime.h>
#include <stdint.h>

#define N_LEVELS   16
#define HASH_SIZE  (1 << 19)
#define HASH_MASK  ((unsigned)HASH_SIZE - 1u)
#define STAGED_ROWS 12096     // level0 (16^3=4096) + level1 (20^3=8000) rows, 96,768 B
#define BLOCK 256

typedef float f2 __attribute__((ext_vector_type(2)));
typedef float f4 __attribute__((ext_vector_type(4)));
typedef unsigned int u32x4 __attribute__((ext_vector_type(4)));
typedef int i32x8 __attribute__((ext_vector_type(8)));
typedef int i32x4 __attribute__((ext_vector_type(4)));

struct LevelMeta {
  int res[N_LEVELS];
  int row_off[N_LEVELS];
  int dense[N_LEVELS];
};

__global__ __launch_bounds__(BLOCK) void hashgrid_fwd_kernel(
    const float* __restrict__ x, const float* __restrict__ tables,
    float* __restrict__ out, LevelMeta meta, int npts)
{
  extern __shared__ f2 s_tab[];   // staged rows [0, STAGED_ROWS) of `tables`

  // ---- Stage level-0/1 embedding tables into LDS via the Tensor Data Mover.
#if __has_builtin(__builtin_amdgcn_tensor_load_to_lds)
  if (threadIdx.x < 32) {   // one wave issues the TDM descriptor (EXEC ignored)
    unsigned long long ga = (unsigned long long)(uintptr_t)tables;
    const unsigned nrows = STAGED_ROWS;          // 8-byte elements
    u32x4 g0; i32x8 g1; i32x4 g2; i32x4 g3; i32x8 g4;
    g0[0] = 1u;                                            // count = 1 (valid D#)
    g0[1] = 0u;                                            // lds_addr = 0 (only LDS object)
    g0[2] = (unsigned)(ga & 0xFFFFFFFFu);                  // global_addr[31:0]
    g0[3] = (unsigned)((ga >> 32) & 0x1FFFFFFu)            // global_addr[56:32]
          | (2u << 30);                                    // type = 2 ("image")
    g1[0] = (int)(3u << 16);                               // data_size = 3 (8 bytes)
    g1[1] = (int)((nrows & 0xFFFFu) << 16);                // tensor_dim0[15:0]
    g1[2] = (int)(((nrows >> 16) & 0xFFFFu) | (1u << 16)); // dim0[31:16] | tensor_dim1=1
    g1[3] = (int)((nrows & 0xFFFFu) << 16);                // tile_dim0 = nrows (1D tile)
    g1[4] = 1;                                             // tile_dim1 = 1, tile_dim2 = 0
    g1[5] = (int)nrows;                                    // tensor_dim0_stride[31:0]
    g1[6] = 0; g1[7] = 0;                                  // dim1_stride = 0
    g2[0]=g2[1]=g2[2]=g2[3]=0;                             // group2 unused (<=2D tile)
    g3[0]=g3[1]=g3[2]=g3[3]=0;                             // group3 unused
    g4[0]=g4[1]=g4[2]=g4[3]=g4[4]=g4[5]=g4[6]=g4[7]=0;     // extra operand (6-arg form)
    __builtin_amdgcn_tensor_load_to_lds(g0, g1, g2, g3, g4, 0);
#if __has_builtin(__builtin_amdgcn_s_wait_tensorcnt)
    __builtin_amdgcn_s_wait_tensorcnt(0);
#else
    asm volatile("s_wait_tensorcnt 0x0" ::: "memory");
#endif
  }
#else
  for (unsigned i = threadIdx.x; i < STAGED_ROWS; i += BLOCK)
    s_tab[i] = ((const f2*)tables)[i];
#endif
  __syncthreads();

  int n = blockIdx.x * BLOCK + threadIdx.x;
  if (n >= npts) return;

  // Streaming read of the point (never reused -> non-temporal).
  const float* xp = x + 3ull * (unsigned)n;
  float px = __builtin_nontemporal_load(xp + 0);
  float py = __builtin_nontemporal_load(xp + 1);
  float pz = __builtin_nontemporal_load(xp + 2);
  px = fminf(fmaxf(px, 0.0f), 1.0f);
  py = fminf(fmaxf(py, 0.0f), 1.0f);
  pz = fminf(fmaxf(pz, 0.0f), 1.0f);

  const f2* __restrict__ gtab = (const f2*)tables;
  float acc[2 * N_LEVELS];

#pragma unroll
  for (int lvl = 0; lvl < N_LEVELS; ++lvl) {
    const int   r  = meta.res[lvl];
    const float rf = (float)r;
    // x*r + 0.5 with NO fma-contraction (must match reference rounding at floor()).
    float ox = __fadd_rn(__fmul_rn(px, rf), 0.5f);
    float oy = __fadd_rn(__fmul_rn(py, rf), 0.5f);
    float oz = __fadd_rn(__fmul_rn(pz, rf), 0.5f);
    float fx = floorf(ox), fy = floorf(oy), fz = floorf(oz);
    int bx = (int)fx, by = (int)fy, bz = (int)fz;
    float tx = __fsub_rn(ox, fx), ty = __fsub_rn(oy, fy), tz = __fsub_rn(oz, fz);
    float sx = __fsub_rn(1.0f, tx), sy = __fsub_rn(1.0f, ty), sz = __fsub_rn(1.0f, tz);

    const int base = meta.row_off[lvl];
    const int is_dense = meta.dense[lvl];   // uniform (scalar) branch
    float a0 = 0.0f, a1 = 0.0f;

#pragma unroll
    for (int c = 0; c < 8; ++c) {
      // corner index = 4*i + 2*j + k  -> bit2: x, bit1: y, bit0: z
      int ix = bx + ((c >> 2) & 1);
      int iy = by + ((c >> 1) & 1);
      int iz = bz + (c & 1);
      unsigned row;
      if (is_dense) {
        // vox in [0, r+1]; one conditional subtract == % r here
        int wx = (ix >= r) ? ix - r : ix;
        int wy = (iy >= r) ? iy - r : iy;
        int wz = (iz >= r) ? iz - r : iz;
        row = (unsigned)(wx + r * (wy + r * wz));
      } else {
        row = ((unsigned)ix * 1u)
            ^ ((unsigned)iy * 2654435761u)
            ^ ((unsigned)iz * 805459861u);
        row &= HASH_MASK;
      }
      f2 e;
      if (lvl < 2) e = s_tab[(unsigned)base + row];   // LDS-resident levels (TDM-staged)
      else         e = gtab[(unsigned)base + row];    // L2-resident (tables ~42MB << 192MB L2)
      float w = ((c & 4) ? tx : sx) * ((c & 2) ? ty : sy) * ((c & 1) ? tz : sz);
      a0 += w * e.x;
      a1 += w * e.y;
    }
    acc[2 * lvl]     = a0;
    acc[2 * lvl + 1] = a1;
  }

  // 128 B per point, write-once stream -> non-temporal b128 stores.
  f4* o = (f4*)(out + 32ull * (unsigned)n);
#pragma unroll
  for (int i = 0; i < 8; ++i) {
    f4 v;
    v.x = acc[4 * i + 0]; v.y = acc[4 * i + 1];
    v.z = acc[4 * i + 2]; v.w = acc[4 * i + 3];
    __builtin_nontemporal_store(v, o + i);
  }
}

// ---------------- host side ----------------

static long long total_rows_for(const int* res) {
  long long t = 0;
  for (int l = 0; l < N_LEVELS; ++l) {
    long long c = (long long)res[l] * res[l] * res[l];
    t += (c <= (long long)HASH_SIZE) ? c : (long long)HASH_SIZE;
  }
  return t;
}

static void build_meta(long long total_rows, LevelMeta* m) {
  // growth factor b = fl32(exp((ln512-ln16)/15)) sits within ~1 ulp of 2^(1/3);
  // its rounding direction decides res at the power-of-two levels. Disambiguate
  // via the actual table row count supplied by the harness.
  static const int RES_A[N_LEVELS] =
      {16,20,25,32,40,50,64,80,101,128,161,203,256,322,406,512};
  static const int RES_B[N_LEVELS] =
      {16,20,25,31,40,50,63,80,101,127,161,203,255,322,406,511};
  const int* res = RES_A;
  if (total_rows_for(RES_B) == total_rows) res = RES_B;
  long long off = 0;
  for (int l = 0; l < N_LEVELS; ++l) {
    long long c  = (long long)res[l] * res[l] * res[l];
    long long sz = (c <= (long long)HASH_SIZE) ? c : (long long)HASH_SIZE;
    m->res[l]     = res[l];
    m->row_off[l] = (int)off;
    m->dense[l]   = (c <= (long long)HASH_SIZE) ? 1 : 0;
    off += sz;
  }
}

extern "C" void kernel_launch(void* const* d_in, const int* in_sizes, int n_in,
                              void* d_out, int out_size, void* d_ws, size_t ws_size,
                              hipStream_t stream) {
  const float* x      = (const float*)d_in[0];
  const float* tables = (const float*)d_in[1];
  float*       out    = (float*)d_out;
  int npts = in_sizes[0] / 3;
  long long total_rows = (long long)in_sizes[1] / 2;   // tables is [rows, 2]
  LevelMeta m;
  build_meta(total_rows, &m);
  int blocks = (npts + BLOCK - 1) / BLOCK;
  size_t lds_bytes = (size_t)STAGED_ROWS * sizeof(f2);  // 96,768 B
  hashgrid_fwd_kernel<<<blocks, BLOCK, lds_bytes, stream>>>(x, tables, out, m, npts);
}